// SimpleGraphLearner_45767171506720
// MI455X (gfx1250) — compile-verified
//
#include <hip/hip_runtime.h>
#include <hip/hip_bf16.h>
#include <math.h>

// CDNA5 wave32 WMMA fragment types for V_WMMA_F32_16X16X4_F32:
//   A: 16x4 f32  -> 64 vals / 32 lanes = 2 VGPRs
//   B: 4x16 f32  -> 2 VGPRs
//   C/D: 16x16 f32 -> 8 VGPRs
typedef __attribute__((ext_vector_type(2))) float v2f;
typedef __attribute__((ext_vector_type(8))) float v8f;

__device__ __forceinline__ v8f wmma_f32_k4(v2f a, v2f b, v8f c) {
    // 8 args: (neg_a, A, neg_b, B, c_mod, C, reuse_a, reuse_b)
    return __builtin_amdgcn_wmma_f32_16x16x4_f32(false, a, false, b, (short)0, c,
                                                 false, false);
}

// -----------------------------------------------------------------------------
// Kernel 1/2: fused encoder + mean over batch.
//   out[n,k] = (1/B) * sum_b relu( sum_j relu(data[b,n]*W0[j]+b0[j]) * W1[j,k] + b1[k] )
// One workgroup (8 waves, 256 threads) per n. h is regenerated on the fly from
// the scalar data[b,n] (2 VFMA+relu per A element) -- never touches HBM.
// Mean-over-b reduced with ds_add_f32 into shared accumulator.
// -----------------------------------------------------------------------------
template <int H>
__global__ __launch_bounds__(256) void encoder_mean_kernel(
    const float* __restrict__ data,  // B x N
    const float* __restrict__ W0,    // H
    const float* __restrict__ b0,    // H
    const float* __restrict__ W1,    // H x H  (read via L2/WGP$, 16-64KB hot)
    const float* __restrict__ b1,    // H
    float* __restrict__ outF,        // N x H
    int Bsz, int Nsz) {
    constexpr int NT = H / 16;  // column tiles per 16-row block
    constexpr int KS = H / 4;   // K steps of 4

    __shared__ float sW0[H];
    __shared__ float sb0[H];
    __shared__ float sb1[H];
    __shared__ float sacc[H];

    const int n = blockIdx.x;
    const int tid = threadIdx.x;

    if (tid < H) {
        sW0[tid] = W0[tid];
        sb0[tid] = b0[tid];
        sb1[tid] = b1[tid];
        sacc[tid] = 0.0f;
    }
    __syncthreads();

    const int wave = tid >> 5;   // wave32
    const int lane = tid & 31;
    const int M = lane & 15;     // row within tile / col within B frag
    const int half = lane >> 4;  // K-pair selector (0 -> K0,K1 ; 1 -> K2,K3)

    // 16 row-blocks of b, 8 waves -> 2 blocks per wave
    for (int rb = wave; rb < (Bsz >> 4); rb += 8) {
        const int row0 = rb * 16;
        const float dval = data[(size_t)(row0 + M) * Nsz + n];

        v8f acc[NT];
#pragma unroll
        for (int t = 0; t < NT; ++t) acc[t] = (v8f){};

#pragma unroll 4
        for (int ks = 0; ks < KS; ++ks) {
            const int kb = ks * 4 + half * 2;
            v2f a;
            a.x = fmaxf(dval * sW0[kb] + sb0[kb], 0.0f);
            a.y = fmaxf(dval * sW0[kb + 1] + sb0[kb + 1], 0.0f);
#pragma unroll
            for (int t = 0; t < NT; ++t) {
                const int col = t * 16 + M;
                v2f bf;
                bf.x = W1[kb * H + col];
                bf.y = W1[(kb + 1) * H + col];
                acc[t] = wmma_f32_k4(a, bf, acc[t]);
            }
        }

        // bias + relu + partial row-sum (each lane owns 8 rows of the tile),
        // then LDS float atomic to accumulate the batch mean.
#pragma unroll
        for (int t = 0; t < NT; ++t) {
            const int col = t * 16 + M;
            const float bias = sb1[col];
            float s = 0.0f;
#pragma unroll
            for (int r = 0; r < 8; ++r) s += fmaxf(acc[t][r] + bias, 0.0f);
            atomicAdd(&sacc[col], s);
        }
    }
    __syncthreads();

    if (tid < H) outF[(size_t)n * H + tid] = sacc[tid] * (1.0f / (float)Bsz);
}

// -----------------------------------------------------------------------------
// Kernel 3: left/right edge projections via fp32 WMMA.
//   combined[n] = [nf[n] (128) | itf[n] (64)]  (192)
//   left  = combined @ W_edge[:192]   (512x128)
//   right = combined @ W_edge[192:]   (512x128)
// One workgroup per 16-row block; wave w owns column tile w for BOTH outputs.
// -----------------------------------------------------------------------------
__global__ __launch_bounds__(256) void edge_proj_kernel(
    const float* __restrict__ nf,     // 512 x 128
    const float* __restrict__ itf,    // 512 x 64
    const float* __restrict__ Wedge,  // 384 x 128
    float* __restrict__ left,         // 512 x 128
    float* __restrict__ right) {      // 512 x 128
    constexpr int CD = 192;
    __shared__ float scomb[16 * CD];  // 12 KB

    const int n0 = blockIdx.x * 16;
    const int tid = threadIdx.x;

    for (int i = tid; i < 16 * CD; i += 256) {
        const int r = i / CD;
        const int j = i - r * CD;
        scomb[i] = (j < 128) ? nf[(size_t)(n0 + r) * 128 + j]
                             : itf[(size_t)(n0 + r) * 64 + (j - 128)];
    }
    __syncthreads();

    const int wave = tid >> 5;
    const int lane = tid & 31;
    const int M = lane & 15;
    const int half = lane >> 4;
    const int col = wave * 16 + M;

    v8f accL = (v8f){};
    v8f accR = (v8f){};

#pragma unroll 4
    for (int ks = 0; ks < CD / 4; ++ks) {
        const int kb = ks * 4 + half * 2;
        v2f a;
        a.x = scomb[M * CD + kb];
        a.y = scomb[M * CD + kb + 1];
        v2f bl, br;
        bl.x = Wedge[kb * 128 + col];
        bl.y = Wedge[(kb + 1) * 128 + col];
        br.x = Wedge[(CD + kb) * 128 + col];
        br.y = Wedge[(CD + kb + 1) * 128 + col];
        accL = wmma_f32_k4(a, bl, accL);
        accR = wmma_f32_k4(a, br, accR);
    }

#pragma unroll
    for (int r = 0; r < 8; ++r) {
        const int row = n0 + r + half * 8;  // D layout: lanes 0-15 M=r, 16-31 M=r+8
        left[(size_t)row * 128 + col] = accL[r];
        right[(size_t)row * 128 + col] = accR[r];
    }
}

// -----------------------------------------------------------------------------
// Kernel 4: pairwise edge scores (relu inside the h-reduction is not
// factorable, so this is a straight VALU reduction; 512*512*128 fused ops).
//   out[i,j] = (i==j) ? 0.5 : sigmoid( sum_h relu(L[i,h]+R[j,h]+be[h])*wf[h] + bf )
// -----------------------------------------------------------------------------
__global__ __launch_bounds__(256) void edge_logits_kernel(
    const float* __restrict__ left, const float* __restrict__ right,
    const float* __restrict__ b_edge, const float* __restrict__ W_final,
    const float* __restrict__ b_final, float* __restrict__ out, int Nn) {
    __shared__ float sL[128];
    __shared__ float sBe[128];
    __shared__ float sWf[128];

    const int i = blockIdx.x;
    const int tid = threadIdx.x;
    if (tid < 128) {
        sL[tid] = left[(size_t)i * 128 + tid];
        sBe[tid] = b_edge[tid];
        sWf[tid] = W_final[tid];
    }
    __syncthreads();

    const float bf = b_final[0];
    for (int j = tid; j < Nn; j += blockDim.x) {
        const float* rj = right + (size_t)j * 128;
        float acc = 0.0f;
#pragma unroll 8
        for (int h = 0; h < 128; ++h)
            acc += fmaxf(sL[h] + rj[h] + sBe[h], 0.0f) * sWf[h];
        const float logit = acc + bf;
        out[(size_t)i * Nn + j] =
            (j == i) ? 0.5f : (1.0f / (1.0f + __expf(-logit)));
    }
}

// -----------------------------------------------------------------------------
extern "C" void kernel_launch(void* const* d_in, const int* in_sizes, int n_in,
                              void* d_out, int out_size, void* d_ws,
                              size_t ws_size, hipStream_t stream) {
    const float* data   = (const float*)d_in[0];
    const float* mask   = (const float*)d_in[1];
    const float* W_enc0 = (const float*)d_in[2];
    const float* b_enc0 = (const float*)d_in[3];
    const float* W_enc1 = (const float*)d_in[4];
    const float* b_enc1 = (const float*)d_in[5];
    const float* W_int0 = (const float*)d_in[6];
    const float* b_int0 = (const float*)d_in[7];
    const float* W_int1 = (const float*)d_in[8];
    const float* b_int1 = (const float*)d_in[9];
    const float* W_edge = (const float*)d_in[10];
    const float* b_edge = (const float*)d_in[11];
    const float* W_final = (const float*)d_in[12];
    const float* b_final = (const float*)d_in[13];
    float* out = (float*)d_out;

    constexpr int N = 512, B = 256, H = 128, HI = 64;

    // Workspace layout (floats): nf[512*128] | itf[512*64] | left[512*128] | right[512*128]
    float* nf = (float*)d_ws;
    float* itf = nf + (size_t)N * H;
    float* left = itf + (size_t)N * HI;
    float* right = left + (size_t)N * H;

    encoder_mean_kernel<H><<<N, 256, 0, stream>>>(data, W_enc0, b_enc0, W_enc1,
                                                  b_enc1, nf, B, N);
    encoder_mean_kernel<HI><<<N, 256, 0, stream>>>(mask, W_int0, b_int0, W_int1,
                                                   b_int1, itf, B, N);
    edge_proj_kernel<<<N / 16, 256, 0, stream>>>(nf, itf, W_edge, left, right);
    edge_logits_kernel<<<N, 256, 0, stream>>>(left, right, b_edge, W_final,
                                              b_final, out, N);
}